// TransformerEncoder_72696616452676
// MI455X (gfx1250) — compile-verified
//
#include <hip/hip_runtime.h>
#include <math.h>

#define BATCH 4
#define D     240
#define FF    1024
#define NH    8
#define DH    30
#define SVALID 1094
#define SPAD   1104
#define NM1    515
#define TEND   1030

typedef __attribute__((ext_vector_type(2))) float v2f;
typedef __attribute__((ext_vector_type(8))) float v8f;
typedef int v4i __attribute__((vector_size(16)));   // matches builtin prototype

#if defined(__gfx1250__) && __has_builtin(__builtin_amdgcn_global_load_async_to_lds_b128)
#define HAVE_ASYNC_LDS 1
#else
#define HAVE_ASYNC_LDS 0
#endif

__device__ __forceinline__ void wait_async0() {
#if __has_builtin(__builtin_amdgcn_s_wait_asynccnt)
  __builtin_amdgcn_s_wait_asynccnt(0);
#else
  asm volatile("s_wait_asynccnt 0x0" ::: "memory");
#endif
}

__device__ __forceinline__ v8f wmma4(v2f a, v2f b, v8f c) {
  // D = A(16x4) * B(4x16) + C, fp32 everywhere
  return __builtin_amdgcn_wmma_f32_16x16x4_f32(false, a, false, b, (short)0, c, false, false);
}

// ---------------------------------------------------------------------------
// Token assembly: build src[B][SPAD][D] and coords[B][SPAD][3]
// ---------------------------------------------------------------------------
__global__ void assemble_kernel(
    const float* __restrict__ ht_t,  const float* __restrict__ hd_t,
    const float* __restrict__ ht_m1, const float* __restrict__ hd_m1,
    const float* __restrict__ c_ht_t,  const float* __restrict__ c_hd_t,
    const float* __restrict__ c_ht_m1, const float* __restrict__ c_hd_m1,
    const float* __restrict__ st_t, const float* __restrict__ st_m1,
    const float* __restrict__ hp_m1, const float* __restrict__ dp_m1,
    const float* __restrict__ hp_t,  const float* __restrict__ dp_t,
    const float* __restrict__ perc,
    float* __restrict__ src, float* __restrict__ crd)
{
  const int row = blockIdx.x;
  const int b = row / SPAD, s = row % SPAD;
  const int t = threadIdx.x;
  const float* sp = nullptr;
  const float* cp = nullptr;
  if      (s == 0)   sp = st_m1 + (size_t)b * D;
  else if (s == 1)   sp = hp_m1 + (size_t)b * D;
  else if (s == 2)   sp = dp_m1 + (size_t)b * D;
  else if (s < 259)  { sp = ht_m1 + ((size_t)b*256 + (s-3))*D;   cp = c_ht_m1 + ((size_t)b*256 + (s-3))*3; }
  else if (s < 515)  { sp = hd_m1 + ((size_t)b*256 + (s-259))*D; cp = c_hd_m1 + ((size_t)b*256 + (s-259))*3; }
  else if (s == 515) sp = st_t + (size_t)b * D;
  else if (s == 516) sp = hp_t + (size_t)b * D;
  else if (s == 517) sp = dp_t + (size_t)b * D;
  else if (s < 774)  { sp = ht_t + ((size_t)b*256 + (s-518))*D;  cp = c_ht_t + ((size_t)b*256 + (s-518))*3; }
  else if (s < 1030) { sp = hd_t + ((size_t)b*256 + (s-774))*D;  cp = c_hd_t + ((size_t)b*256 + (s-774))*3; }
  else if (s < SVALID) sp = perc + ((size_t)b*64 + (s-1030))*D;
  const size_t base = (size_t)row * D;
  if (t < D) src[base + t] = sp ? sp[t] : 0.f;
  if (t < 3) crd[(size_t)row*3 + t] = cp ? cp[t] : 0.f;
}

// ---------------------------------------------------------------------------
// Generic tiled GEMM:  Y[b][s][n] = sum_k X[b][s][k] * W[n][k] + bias[n]
// X: [B][SPAD][K] contiguous, W: [N][K], Y: [B][SPAD][N]
// One wave per block; 16xK strip of X staged in LDS via CDNA5 async-to-LDS
// (ASYNCcnt) when available; ntPerBlock n-tiles/block.
// act: 0 = none, 1 = exact GELU.
// ---------------------------------------------------------------------------
__global__ __launch_bounds__(32)
void gemm_bias_kernel(const float* __restrict__ X, const float* __restrict__ W,
                      const float* __restrict__ bias, float* __restrict__ Y,
                      int K, int N, int ntPerBlock, int act)
{
  const int b    = blockIdx.z;
  const int s0   = blockIdx.x << 4;
  const int lane = threadIdx.x;
  const int m    = lane & 15;
  const int kb   = (lane >> 4) << 1;   // 0 or 2 (K half per A/B layout)
  const int n    = lane & 15;
  const int mrow = (lane >> 4) << 3;   // 0 or 8 (C/D row block)

  extern __shared__ float ldsX[];      // [16][K]
  const float* Xb = X + ((size_t)b * SPAD + s0) * (size_t)K;
  const int nvec = (16 * K) >> 2;      // float4 count (exact multiple of 32)
#if HAVE_ASYNC_LDS
  {
    typedef __attribute__((address_space(1))) v4i* gp;
    typedef __attribute__((address_space(3))) v4i* lp;
    gp gsrc = (gp)const_cast<float*>(Xb);
    lp ldst = (lp)ldsX;
    for (int i = lane; i < nvec; i += 32)
      __builtin_amdgcn_global_load_async_to_lds_b128(gsrc + i, ldst + i, 0, 0);
    wait_async0();
  }
#else
  for (int i = lane; i < nvec; i += 32)
    ((float4*)ldsX)[i] = ((const float4*)Xb)[i];
#endif
  __syncthreads();

  float* Yb = Y + (size_t)b * SPAD * (size_t)N;
  const int nt0 = blockIdx.y * ntPerBlock;
  const float* Ar = ldsX + m * K + kb;
  for (int tt = 0; tt < ntPerBlock; ++tt) {
    const int nt = (nt0 + tt) << 4;
    if (nt >= N) break;
    v8f c = {0.f,0.f,0.f,0.f,0.f,0.f,0.f,0.f};
    const float* Wr = W + (size_t)(nt + n) * (size_t)K + kb;
    // speculative prefetch of next n-tile's W rows (global_prefetch_b8)
    __builtin_prefetch(Wr + (size_t)16 * K, 0, 0);
    for (int kk = 0; kk < K; kk += 4) {
      v2f a;  a.x  = Ar[kk]; a.y  = Ar[kk + 1];
      v2f bb; bb.x = Wr[kk]; bb.y = Wr[kk + 1];
      c = wmma4(a, bb, c);
    }
    const float bv = bias[nt + n];
#pragma unroll
    for (int vv = 0; vv < 8; ++vv) {
      float val = c[vv] + bv;
      if (act) val = 0.5f * val * (1.f + erff(val * 0.70710678118654752f));
      Yb[(size_t)(s0 + mrow + vv) * N + (nt + n)] = val;
    }
  }
}

// ---------------------------------------------------------------------------
// 3-axis rotary applied in place to q and k (layout [B][SPAD][D])
// ---------------------------------------------------------------------------
__global__ void rotary_kernel(float* __restrict__ q, float* __restrict__ k,
                              const float* __restrict__ crd)
{
  const int row = blockIdx.x;                 // b * SVALID + s
  const int b = row / SVALID, s = row % SVALID;
  const int t = threadIdx.x;
  if (t >= NH * 3 * 5) return;                // 8 heads * 3 axes * 5 pairs
  const int h = t / 15, r = t % 15, a = r / 5, p = r % 5;
  const float inv = __powf(10000.f, -(float)p / 5.f);
  const float ang = crd[((size_t)b * SPAD + s) * 3 + a] * inv;
  const float cs = cosf(ang), sn = sinf(ang);
  const size_t base = ((size_t)b * SPAD + s) * D + h * DH + a * 10 + 2 * p;
  float x1 = q[base], x2 = q[base + 1];
  q[base]     = x1 * cs - x2 * sn;
  q[base + 1] = x1 * sn + x2 * cs;
  x1 = k[base]; x2 = k[base + 1];
  k[base]     = x1 * cs - x2 * sn;
  k[base + 1] = x1 * sn + x2 * cs;
}

// ---------------------------------------------------------------------------
// Attention: one wave per (b, head, 16-row q-tile).
// Scores in LDS [16][SPAD] via WMMA, masked softmax, then P*V via WMMA.
// ---------------------------------------------------------------------------
__global__ __launch_bounds__(32)
void attn_kernel(const float* __restrict__ Q, const float* __restrict__ Km,
                 const float* __restrict__ V, float* __restrict__ O)
{
  extern __shared__ float P[];                // [16][SPAD]
  const int b = blockIdx.z, h = blockIdx.y;
  const int s0 = blockIdx.x << 4;
  const int lane = threadIdx.x;
  const int m    = lane & 15;
  const int kb   = (lane >> 4) << 1;
  const int n    = lane & 15;
  const int mrow = (lane >> 4) << 3;
  const int hc = h * DH;
  const float* Qb = Q + (size_t)b * SPAD * D;
  const float* Kb = Km + (size_t)b * SPAD * D;
  const float* Vb = V + (size_t)b * SPAD * D;

  // Preload this lane's q fragment (Dh=30 zero-padded to 32).
  float areg[16];
#pragma unroll
  for (int i = 0; i < 8; ++i) {
    const int k0 = 4 * i + kb;
    areg[2*i]   = (k0     < DH) ? Qb[(size_t)(s0 + m) * D + hc + k0]     : 0.f;
    areg[2*i+1] = (k0 + 1 < DH) ? Qb[(size_t)(s0 + m) * D + hc + k0 + 1] : 0.f;
  }

  // Phase 1: scores -> LDS with scale + mask
  for (int kt = 0; kt < SPAD; kt += 16) {
    v8f c = {0.f,0.f,0.f,0.f,0.f,0.f,0.f,0.f};
#pragma unroll
    for (int i = 0; i < 8; ++i) {
      const int k0 = 4 * i + kb;
      v2f a;  a.x = areg[2*i]; a.y = areg[2*i+1];
      v2f bb;
      bb.x = (k0     < DH) ? Kb[(size_t)(kt + n) * D + hc + k0]     : 0.f;
      bb.y = (k0 + 1 < DH) ? Kb[(size_t)(kt + n) * D + hc + k0 + 1] : 0.f;
      c = wmma4(a, bb, c);
    }
    const int kcol = kt + n;
#pragma unroll
    for (int vv = 0; vv < 8; ++vv) {
      const int qrow = s0 + mrow + vv;
      const float sc = c[vv] * 0.18257418583505536f;   // 1/sqrt(30)
      const bool masked =
          (kcol >= SVALID) ||
          (qrow < NM1 && kcol >= NM1 && kcol < TEND) ||
          (qrow >= TEND && qrow < SVALID && kcol < TEND);
      P[(mrow + vv) * SPAD + kcol] = masked ? -__builtin_inff() : sc;
    }
  }
  __syncthreads();

  // Phase 2: row-wise softmax (wave-cooperative per row)
  for (int r = 0; r < 16; ++r) {
    float mx = -__builtin_inff();
    for (int i = lane; i < SPAD; i += 32) mx = fmaxf(mx, P[r * SPAD + i]);
#pragma unroll
    for (int off = 16; off; off >>= 1) mx = fmaxf(mx, __shfl_xor(mx, off, 32));
    float sum = 0.f;
    for (int i = lane; i < SPAD; i += 32) {
      const float e = expf(P[r * SPAD + i] - mx);
      P[r * SPAD + i] = e;
      sum += e;
    }
#pragma unroll
    for (int off = 16; off; off >>= 1) sum += __shfl_xor(sum, off, 32);
    const float inv = 1.f / sum;
    for (int i = lane; i < SPAD; i += 32) P[r * SPAD + i] *= inv;
  }
  __syncthreads();

  // Phase 3: O = P * V  (two 16-wide n-tiles cover Dh=30)
  for (int nt = 0; nt < 32; nt += 16) {
    const int col = nt + n;
    const bool cok = (col < DH);
    v8f c = {0.f,0.f,0.f,0.f,0.f,0.f,0.f,0.f};
    for (int kk = 0; kk < SPAD; kk += 4) {
      v2f a;  a.x = P[m * SPAD + kk + kb]; a.y = P[m * SPAD + kk + kb + 1];
      v2f bb;
      bb.x = cok ? Vb[(size_t)(kk + kb)     * D + hc + col] : 0.f;
      bb.y = cok ? Vb[(size_t)(kk + kb + 1) * D + hc + col] : 0.f;
      c = wmma4(a, bb, c);
    }
    if (cok) {
#pragma unroll
      for (int vv = 0; vv < 8; ++vv)
        O[((size_t)b * SPAD + s0 + mrow + vv) * D + hc + col] = c[vv];
    }
  }
}

// ---------------------------------------------------------------------------
// Out = LayerNorm(X + Rsd) * g + be ; one wave per row; pad rows -> 0
// ---------------------------------------------------------------------------
__global__ __launch_bounds__(32)
void add_ln_kernel(const float* __restrict__ X, const float* __restrict__ Rsd,
                   const float* __restrict__ g, const float* __restrict__ be,
                   float* __restrict__ Out)
{
  const int row = blockIdx.x;
  const int s = row % SPAD;
  const int lane = threadIdx.x;
  const size_t base = (size_t)row * D;
  if (s >= SVALID) {
    for (int i = lane; i < D; i += 32) Out[base + i] = 0.f;
    return;
  }
  float vals[8];
  float mean = 0.f;
#pragma unroll
  for (int i = 0; i < 8; ++i) {
    const int c = lane + i * 32;
    const float x = (c < D) ? (X[base + c] + Rsd[base + c]) : 0.f;
    vals[i] = x; mean += x;
  }
#pragma unroll
  for (int off = 16; off; off >>= 1) mean += __shfl_xor(mean, off, 32);
  mean *= (1.f / D);
  float var = 0.f;
#pragma unroll
  for (int i = 0; i < 8; ++i) {
    const int c = lane + i * 32;
    if (c < D) { const float d = vals[i] - mean; var += d * d; }
  }
#pragma unroll
  for (int off = 16; off; off >>= 1) var += __shfl_xor(var, off, 32);
  var *= (1.f / D);
  const float rstd = rsqrtf(var + 1e-5f);
#pragma unroll
  for (int i = 0; i < 8; ++i) {
    const int c = lane + i * 32;
    if (c < D) Out[base + c] = (vals[i] - mean) * rstd * g[c] + be[c];
  }
}

// ---------------------------------------------------------------------------
// Pack valid rows [B][SVALID][D] -> d_out
// ---------------------------------------------------------------------------
__global__ void pack_kernel(const float* __restrict__ src, float* __restrict__ out)
{
  const int row = blockIdx.x;                 // b * SVALID + s
  const int b = row / SVALID, s = row % SVALID;
  const int t = threadIdx.x;
  if (t < D) out[(size_t)row * D + t] = src[((size_t)b * SPAD + s) * D + t];
}

// ---------------------------------------------------------------------------
extern "C" void kernel_launch(void* const* d_in, const int* in_sizes, int n_in,
                              void* d_out, int out_size, void* d_ws, size_t ws_size,
                              hipStream_t stream)
{
  (void)in_sizes; (void)n_in; (void)out_size;
  const float* HT_T  = (const float*)d_in[0];
  const float* HD_T  = (const float*)d_in[1];
  const float* HT_M1 = (const float*)d_in[2];
  const float* HD_M1 = (const float*)d_in[3];
  const float* C_HT_T  = (const float*)d_in[4];
  const float* C_HD_T  = (const float*)d_in[5];
  const float* C_HT_M1 = (const float*)d_in[6];
  const float* C_HD_M1 = (const float*)d_in[7];
  const float* ST_T  = (const float*)d_in[8];
  const float* ST_M1 = (const float*)d_in[9];
  const float* HP_M1 = (const float*)d_in[10];
  const float* DP_M1 = (const float*)d_in[11];
  const float* HP_T  = (const float*)d_in[12];
  const float* DP_T  = (const float*)d_in[13];
  const float* PERC  = (const float*)d_in[14];

  const size_t SZ_D = (size_t)BATCH * SPAD * D;
  const size_t SZ_C = (size_t)BATCH * SPAD * 3;
  const size_t SZ_F = (size_t)BATCH * SPAD * FF;
  float* ws    = (float*)d_ws;
  float* src   = ws;
  float* crd   = src  + SZ_D;
  float* qraw  = crd  + SZ_C;
  float* kraw  = qraw + SZ_D;
  float* vraw  = kraw + SZ_D;
  float* aout  = vraw + SZ_D;
  float* oproj = aout + SZ_D;
  float* xbuf  = oproj+ SZ_D;
  float* ff1   = xbuf + SZ_D;
  float* ff2   = ff1  + SZ_F;
  const size_t need = (size_t)(ff2 + SZ_D - ws) * sizeof(float);
  if (ws_size < need) return;

  assemble_kernel<<<BATCH * SPAD, 256, 0, stream>>>(
      HT_T, HD_T, HT_M1, HD_M1, C_HT_T, C_HD_T, C_HT_M1, C_HD_M1,
      ST_T, ST_M1, HP_M1, DP_M1, HP_T, DP_T, PERC, src, crd);

  const size_t shD = (size_t)16 * D  * sizeof(float);
  const size_t shF = (size_t)16 * FF * sizeof(float);
  const size_t shP = (size_t)16 * SPAD * sizeof(float);
  const dim3 gD (SPAD / 16, 5,  BATCH);   // 15 n-tiles / 3 per block
  const dim3 gF1(SPAD / 16, 16, BATCH);   // 64 n-tiles / 4 per block
  const dim3 gA (SPAD / 16, NH, BATCH);

  for (int L = 0; L < 4; ++L) {
    const int pb = 15 + L * 16;
    const float* Wq = (const float*)d_in[pb + 0];
    const float* bq = (const float*)d_in[pb + 1];
    const float* Wk = (const float*)d_in[pb + 2];
    const float* bk = (const float*)d_in[pb + 3];
    const float* Wv = (const float*)d_in[pb + 4];
    const float* bv = (const float*)d_in[pb + 5];
    const float* Wo = (const float*)d_in[pb + 6];
    const float* bo = (const float*)d_in[pb + 7];
    const float* W1 = (const float*)d_in[pb + 8];
    const float* b1 = (const float*)d_in[pb + 9];
    const float* W2 = (const float*)d_in[pb + 10];
    const float* b2 = (const float*)d_in[pb + 11];
    const float* g1 = (const float*)d_in[pb + 12];
    const float* e1 = (const float*)d_in[pb + 13];
    const float* g2 = (const float*)d_in[pb + 14];
    const float* e2 = (const float*)d_in[pb + 15];

    gemm_bias_kernel<<<gD, 32, shD, stream>>>(src, Wq, bq, qraw, D, D, 3, 0);
    gemm_bias_kernel<<<gD, 32, shD, stream>>>(src, Wk, bk, kraw, D, D, 3, 0);
    gemm_bias_kernel<<<gD, 32, shD, stream>>>(src, Wv, bv, vraw, D, D, 3, 0);
    rotary_kernel<<<BATCH * SVALID, 128, 0, stream>>>(qraw, kraw, crd);
    attn_kernel<<<gA, 32, shP, stream>>>(qraw, kraw, vraw, aout);
    gemm_bias_kernel<<<gD, 32, shD, stream>>>(aout, Wo, bo, oproj, D, D, 3, 0);
    add_ln_kernel<<<BATCH * SPAD, 32, 0, stream>>>(oproj, src, g1, e1, xbuf);
    gemm_bias_kernel<<<gF1, 32, shD, stream>>>(xbuf, W1, b1, ff1, D, FF, 4, 1);
    gemm_bias_kernel<<<gD, 32, shF, stream>>>(ff1, W2, b2, ff2, FF, D, 3, 0);
    add_ln_kernel<<<BATCH * SPAD, 32, 0, stream>>>(ff2, xbuf, g2, e2, src);
  }

  pack_kernel<<<BATCH * SVALID, 256, 0, stream>>>(src, (float*)d_out);
}